// BornMultivariateNormalDistribution_49563922595976
// MI455X (gfx1250) — compile-verified
//
#include <hip/hip_runtime.h>
#include <math.h>

// ---------------------------------------------------------------------------
// MVN log-prob on MI455X (gfx1250), f32 WMMA path + async global->LDS copies.
//   B=4096 samples, K=256 components, D=64.
//   Kernel 1 (per k): cov = W W^T (WMMA f32), chol, M = L^-1, hld, v = M*mu.
//   Kernel 2: Y = X * M^T (WMMA f32), maha[b] = sum_n (Y[b,n] - v[n])^2,
//             lp = -0.5*(D*log(2pi) + maha) - hld.   Output complex64.
// ---------------------------------------------------------------------------

typedef float v2f __attribute__((ext_vector_type(2)));
typedef float v8f __attribute__((ext_vector_type(8)));

// Types for the async-LDS builtin (arg0 per hipcc diagnostic:
// 'int __attribute__((vector_size(16))) __device__ *').
typedef int v4i __attribute__((vector_size(16)));
typedef __attribute__((address_space(1))) v4i* as1_v4i_p;
typedef __attribute__((address_space(3))) v4i* as3_v4i_p;

#define WMMA_F32_16x16x4(a, b, c)                                             \
  __builtin_amdgcn_wmma_f32_16x16x4_f32(false, (a), false, (b), (short)0,     \
                                        (c), false, false)

#define KCOMP 256
#define BROWS 4096
#define LDSTRIDE 68 /* 272B rows: 16B-aligned for b128, bank-conflict-free */
#define D_LOG_TWO_PI 117.62413225019809f /* 64 * log(2*pi) */
#define EPS 1e-5f

// --- CDNA5 async global->LDS copy of a 64x64 f32 tile (row stride 64 ->
// --- padded LDS stride 68).  1024 x b128 chunks, 4 per thread (256 thr).
__device__ __forceinline__ void copy_tile_async(const float* __restrict__ g,
                                                float* lds0, int tid) {
#pragma unroll
  for (int j = 0; j < 4; ++j) {
    const int c = tid + j * 256;
    const int row = c >> 4, c4 = (c & 15) * 4;
    const float* gp = g + row * 64 + c4;
    float* lp = lds0 + row * LDSTRIDE + c4;
#if __has_builtin(__builtin_amdgcn_global_load_async_to_lds_b128)
    __builtin_amdgcn_global_load_async_to_lds_b128((as1_v4i_p)gp,
                                                   (as3_v4i_p)lp, 0, 0);
#else
    ((float4*)lp)[0] = ((const float4*)gp)[0];
#endif
  }
}

__device__ __forceinline__ void wait_async_lds() {
#if __has_builtin(__builtin_amdgcn_s_wait_asynccnt)
  __builtin_amdgcn_s_wait_asynccnt(0);
#else
  asm volatile("s_wait_asynccnt 0" ::: "memory");
#endif
}

// ---------------------------------------------------------------------------
// Kernel 1: per-component factorization.  One block per k, 8 wave32.
// ---------------------------------------------------------------------------
__global__ __launch_bounds__(256) void mvn_factor_kernel(
    const float* __restrict__ weight,  // [K, 64, 64]
    const float* __restrict__ mu,      // [K, 64]
    float* __restrict__ Minv,          // [K, 64, 64] out (L^-1, lower tri)
    float* __restrict__ hld,           // [K] out (sum log diag L)
    float* __restrict__ vk)            // [K, 64] out (M * mu)
{
  const int k = blockIdx.x;
  const int tid = threadIdx.x;

  __shared__ float Ws[64][LDSTRIDE];
  __shared__ float Cs[64][LDSTRIDE];  // cov -> L in place
  __shared__ float Mi[64][LDSTRIDE];  // L^-1

  copy_tile_async(weight + (size_t)k * 4096, &Ws[0][0], tid);
  wait_async_lds();
  __syncthreads();

  const int wave = tid >> 5;
  const int lane = tid & 31;
  const int half = lane >> 4;  // 0: K pair {0,1}, 1: K pair {2,3}
  const int l = lane & 15;
  const int tN = wave & 3;
  const int tM0 = wave >> 2;   // tiles (tM0, tN) and (tM0+2, tN): B shared
  const int tM1 = tM0 + 2;

  // cov = W W^T : B fragment loaded once per K-step, feeds two WMMAs.
  {
    v8f acc0 = {}, acc1 = {};
#pragma unroll
    for (int s = 0; s < 16; ++s) {
      const int x0 = s * 4 + half * 2;
      v2f a0, a1, b;
      b.x = Ws[tN * 16 + l][x0];       // B[x][u] = W[u][x]
      b.y = Ws[tN * 16 + l][x0 + 1];
      a0.x = Ws[tM0 * 16 + l][x0];
      a0.y = Ws[tM0 * 16 + l][x0 + 1];
      a1.x = Ws[tM1 * 16 + l][x0];
      a1.y = Ws[tM1 * 16 + l][x0 + 1];
      acc0 = WMMA_F32_16x16x4(a0, b, acc0);
      acc1 = WMMA_F32_16x16x4(a1, b, acc1);
    }
#pragma unroll
    for (int r = 0; r < 8; ++r) {
      Cs[tM0 * 16 + half * 8 + r][tN * 16 + l] = acc0[r];
      Cs[tM1 * 16 + half * 8 + r][tN * 16 + l] = acc1[r];
    }
  }
  __syncthreads();
  if (tid < 64) Cs[tid][tid] += EPS;

  // In-LDS Cholesky (left-looking; only lower triangle touched).
  for (int j = 0; j < 64; ++j) {
    __syncthreads();
    if (tid == 0) {
      float s = Cs[j][j];
      for (int p = 0; p < j; ++p) s -= Cs[j][p] * Cs[j][p];
      Cs[j][j] = sqrtf(fmaxf(s, 1e-30f));
    }
    __syncthreads();
    const int i = j + 1 + tid;
    if (i < 64) {
      float s = Cs[i][j];
      for (int p = 0; p < j; ++p) s -= Cs[i][p] * Cs[j][p];
      Cs[i][j] = s / Cs[j][j];
    }
  }
  __syncthreads();

  // Triangular inverse: one column per thread (forward substitution).
  if (tid < 64) {
    const int c = tid;
    for (int i = 0; i < c; ++i) Mi[i][c] = 0.0f;
    Mi[c][c] = 1.0f / Cs[c][c];
    for (int i = c + 1; i < 64; ++i) {
      float s = 0.0f;
      for (int j = c; j < i; ++j) s += Cs[i][j] * Mi[j][c];
      Mi[i][c] = -s / Cs[i][i];
    }
  }
  __syncthreads();

  float* mo = Minv + (size_t)k * 4096;
  for (int i = tid; i < 4096; i += 256) mo[i] = Mi[i >> 6][i & 63];

  // v = M * mu  (folds the mean subtraction out of kernel 2).
  if (tid < 64) {
    const float* muk = mu + (size_t)k * 64;
    float s = 0.0f;
    for (int j = 0; j <= tid; ++j) s += Mi[tid][j] * muk[j];  // lower tri
    vk[(size_t)k * 64 + tid] = s;
  }
  if (tid == 0) {
    float s = 0.0f;
    for (int i = 0; i < 64; ++i) s += logf(Cs[i][i]);
    hld[k] = s;
  }
}

// ---------------------------------------------------------------------------
// Kernel 2: mahalanobis GEMM + log-prob.  grid = (K, B/64), 8 wave32.
// ---------------------------------------------------------------------------
__global__ __launch_bounds__(256) void mvn_logprob_kernel(
    const float* __restrict__ x,     // [B, 64]
    const float* __restrict__ Minv,  // [K, 64, 64]
    const float* __restrict__ hld,   // [K]
    const float* __restrict__ vk,    // [K, 64]
    float* __restrict__ out)         // [B, K] complex64 interleaved
{
  const int k = blockIdx.x;
  const int bt = blockIdx.y;
  const int tid = threadIdx.x;

  __shared__ float Xs[64][LDSTRIDE];  // raw x tile (mean folded into vk)
  __shared__ float Ms[64][LDSTRIDE];  // M_k
  __shared__ float vS[64];
  __shared__ float maha[64];

  // Pure copies -> CDNA5 async global->LDS (ASYNCcnt), no VGPR round-trip.
  copy_tile_async(x + (size_t)bt * 4096, &Xs[0][0], tid);
  copy_tile_async(Minv + (size_t)k * 4096, &Ms[0][0], tid);
  if (tid < 64) {
    vS[tid] = vk[(size_t)k * 64 + tid];
    maha[tid] = 0.0f;
  }
  wait_async_lds();
  __syncthreads();

  const int wave = tid >> 5;
  const int lane = tid & 31;
  const int half = lane >> 4;
  const int l = lane & 15;
  const int tN = wave & 3;
  const int tM0 = wave >> 2;
  const int tM1 = tM0 + 2;

  // Y = X * M^T ; shared B fragment drives two accumulators per wave.
  v8f acc0 = {}, acc1 = {};
#pragma unroll
  for (int s = 0; s < 16; ++s) {
    const int x0 = s * 4 + half * 2;
    v2f a0, a1, b;
    b.x = Ms[tN * 16 + l][x0];       // B[x][n] = M[n][x]
    b.y = Ms[tN * 16 + l][x0 + 1];
    a0.x = Xs[tM0 * 16 + l][x0];
    a0.y = Xs[tM0 * 16 + l][x0 + 1];
    a1.x = Xs[tM1 * 16 + l][x0];
    a1.y = Xs[tM1 * 16 + l][x0 + 1];
    acc0 = WMMA_F32_16x16x4(a0, b, acc0);
    acc1 = WMMA_F32_16x16x4(a1, b, acc1);
  }

  // Row-wise sum of squares of (Y - v): lane l owns column n = tN*16+l.
  const float vn = vS[tN * 16 + l];
#pragma unroll
  for (int r = 0; r < 8; ++r) {
    const float y0 = acc0[r] - vn;
    const float y1 = acc1[r] - vn;
    atomicAdd(&maha[tM0 * 16 + half * 8 + r], y0 * y0);  // ds_add_f32
    atomicAdd(&maha[tM1 * 16 + half * 8 + r], y1 * y1);
  }
  __syncthreads();

  if (tid < 64) {
    const int b = bt * 64 + tid;
    const float lp = -0.5f * (D_LOG_TWO_PI + maha[tid]) - hld[k];
    const size_t o = (((size_t)b * KCOMP) + k) * 2;
    out[o] = lp;        // real
    out[o + 1] = 0.0f;  // imag (complex64 cast, zero imaginary part)
  }
}

// ---------------------------------------------------------------------------
extern "C" void kernel_launch(void* const* d_in, const int* in_sizes, int n_in,
                              void* d_out, int out_size, void* d_ws,
                              size_t ws_size, hipStream_t stream) {
  const float* x = (const float*)d_in[0];       // [4096, 64]
  const float* mu = (const float*)d_in[1];      // [1,1,256,64]
  const float* weight = (const float*)d_in[2];  // [1,1,256,64,64]

  float* ws = (float*)d_ws;
  float* Minv = ws;                            // K*64*64 floats (4 MB)
  float* hld = Minv + (size_t)KCOMP * 4096;    // K floats
  float* vk = hld + KCOMP;                     // K*64 floats

  mvn_factor_kernel<<<dim3(KCOMP), dim3(256), 0, stream>>>(weight, mu, Minv,
                                                           hld, vk);

  mvn_logprob_kernel<<<dim3(KCOMP, BROWS / 64), dim3(256), 0, stream>>>(
      x, Minv, hld, vk, (float*)d_out);
}